// BlockSparseLinear_13443247636879
// MI455X (gfx1250) — compile-verified
//
#include <hip/hip_runtime.h>
#include <stdint.h>

typedef float v2f __attribute__((ext_vector_type(2)));
typedef float v8f __attribute__((ext_vector_type(8)));

#define TOKENS 4096
#define IN_F   2048
#define OUT_F  2048
#define NBLK   2048
#define TTILE  128   // tokens per workgroup
#define XS     36    // padded LDS row stride (floats): conflict-free, 16B-aligned rows

// gfx1250 async global->LDS copy, 16 bytes per lane, tracked by ASYNCcnt
__device__ __forceinline__ void async_cp16(const float* g, uint32_t lds_byte_addr) {
    asm volatile("global_load_async_to_lds_b128 %0, %1, off"
                 :: "v"(lds_byte_addr), "v"(g) : "memory");
}
__device__ __forceinline__ void wait_async0() {
    asm volatile("s_wait_asynccnt 0x0" ::: "memory");
}

__global__ __launch_bounds__(256)
void BlockSparseLinear_wmma_async(const float* __restrict__ x,
                                  const float* __restrict__ wblk,
                                  const int*   __restrict__ brow,
                                  const int*   __restrict__ bcol,
                                  float*       __restrict__ out)
{
    __shared__ float w_lds[2][32 * XS];      // double-buffered 32x32 weight block
    __shared__ float x_lds[2][TTILE * XS];   // double-buffered 128x32 x tile
    __shared__ int   list[NBLK];             // ordered match list for this row-block
    __shared__ int   cnt[257];

    const int tid  = threadIdx.x;
    const int lane = tid & 31;
    const int wave = tid >> 5;          // 0..7
    const int w2   = (wave & 3) * 2;    // first of two 16-token subtiles
    const int osub = wave >> 2;         // 16-out subtile (0/1)
    const int half = lane >> 4;         // K-pair / M-half select
    const int mlo  = lane & 15;

    const int token0 = blockIdx.x * TTILE;
    const int myrow  = blockIdx.y;

    // ---- phase 0: deterministic, order-preserving match-list build ----
    int local[8], lc = 0;
    #pragma unroll
    for (int j = 0; j < 8; ++j) {
        const int nb = tid * 8 + j;     // contiguous chunk -> global nb order preserved
        if (brow[nb] == myrow) local[lc++] = nb;
    }
    cnt[tid] = lc;
    __syncthreads();
    if (tid == 0) {
        int s = 0;
        for (int k = 0; k < 256; ++k) { const int c = cnt[k]; cnt[k] = s; s += c; }
        cnt[256] = s;
    }
    __syncthreads();
    {
        const int off = cnt[tid];
        for (int j = 0; j < lc; ++j) list[off + j] = local[j];
    }
    __syncthreads();
    const int n = cnt[256];

    // ---- staging layout (256 threads move 4KB weights + 16KB x per block) ----
    const int w_o  = tid >> 3;             // weight row o: 0..31
    const int w_k  = (tid & 7) * 4;        // 16B chunk within row
    const int xt_t = tid >> 1;             // x token: 0..127
    const int xt_k = (tid & 1) * 16;       // half-row select
    const uint32_t wbase[2] = { (uint32_t)(uintptr_t)&w_lds[0][0],
                                (uint32_t)(uintptr_t)&w_lds[1][0] };
    const uint32_t xbase[2] = { (uint32_t)(uintptr_t)&x_lds[0][0],
                                (uint32_t)(uintptr_t)&x_lds[1][0] };

    auto issue = [&](int i, int b) {
        const int nb  = list[i];
        const int col = bcol[nb] * 32;
        async_cp16(wblk + (size_t)nb * 1024 + w_o * 32 + w_k,
                   wbase[b] + (uint32_t)(w_o * XS + w_k) * 4u);
        const float* xs = x + (size_t)(token0 + xt_t) * IN_F + col + xt_k;
        const uint32_t xd = xbase[b] + (uint32_t)(xt_t * XS + xt_k) * 4u;
        async_cp16(xs + 0,  xd + 0);
        async_cp16(xs + 4,  xd + 16);
        async_cp16(xs + 8,  xd + 32);
        async_cp16(xs + 12, xd + 48);
    };

    v8f acc0 = {0.f,0.f,0.f,0.f,0.f,0.f,0.f,0.f};
    v8f acc1 = {0.f,0.f,0.f,0.f,0.f,0.f,0.f,0.f};

    // ---- double-buffered async pipeline ----
    if (n > 0) issue(0, 0);
    for (int i = 0; i < n; ++i) {
        const int b = i & 1;
        wait_async0();        // my copies for block i have landed in LDS
        __syncthreads();      // everyone's landed; all waves done reading buf 1-b (from i-1)
        if (i + 1 < n) issue(i + 1, 1 - b);   // overlaps with compute below

        const float* wl = &w_lds[b][0];
        const float* xl = &x_lds[b][0];
        #pragma unroll
        for (int kk = 0; kk < 32; kk += 4) {
            const v2f bf = *(const v2f*)&wl[(osub * 16 + mlo) * XS + kk + half * 2];
            const v2f a0 = *(const v2f*)&xl[((w2 + 0) * 16 + mlo) * XS + kk + half * 2];
            const v2f a1 = *(const v2f*)&xl[((w2 + 1) * 16 + mlo) * XS + kk + half * 2];
            acc0 = __builtin_amdgcn_wmma_f32_16x16x4_f32(
                false, a0, false, bf, (short)0, acc0, false, false);
            acc1 = __builtin_amdgcn_wmma_f32_16x16x4_f32(
                false, a1, false, bf, (short)0, acc1, false, false);
        }
    }

    // ---- write D: VGPR v -> M = v + half*8, N = mlo ----
    const int ncol = myrow * 32 + osub * 16 + mlo;
    #pragma unroll
    for (int v = 0; v < 8; ++v) {
        const int m0 = token0 + (w2 + 0) * 16 + half * 8 + v;
        const int m1 = token0 + (w2 + 1) * 16 + half * 8 + v;
        out[(size_t)m0 * OUT_F + ncol] = acc0[v];
        out[(size_t)m1 * OUT_F + ncol] = acc1[v];
    }
}

extern "C" void kernel_launch(void* const* d_in, const int* in_sizes, int n_in,
                              void* d_out, int out_size, void* d_ws, size_t ws_size,
                              hipStream_t stream) {
    const float* x    = (const float*)d_in[0];
    const float* wblk = (const float*)d_in[1];
    const int*   brow = (const int*)d_in[2];
    const int*   bcol = (const int*)d_in[3];
    float*       out  = (float*)d_out;

    dim3 grid(TOKENS / TTILE, OUT_F / 32);  // 32 x 64 workgroups
    BlockSparseLinear_wmma_async<<<grid, 256, 0, stream>>>(x, wblk, brow, bcol, out);
}